// SimplestGCNRegress_66675072303276
// MI455X (gfx1250) — compile-verified
//
#include <hip/hip_runtime.h>

typedef __attribute__((ext_vector_type(2))) float v2f;
typedef __attribute__((ext_vector_type(8))) float v8f;

#define F_IN 128
#define F_OUT 16

// ---------------------------------------------------------------- degree init
__global__ __launch_bounds__(256) void k_init_deg(float* __restrict__ deg, int n) {
    int i = blockIdx.x * 256 + threadIdx.x;
    if (i < n) deg[i] = 1.0f;  // self-loop weight
}

// ------------------------------------------------------- deg[col] += w (edges)
__global__ __launch_bounds__(256) void k_deg_scatter(const int* __restrict__ ei,
                                                     const float* __restrict__ ew,
                                                     float* __restrict__ deg, int e) {
    int i = blockIdx.x * 256 + threadIdx.x;
    if (i < e) atomicAdd(&deg[ei[e + i]], ew[i]);  // col = ei[1][i]
}

// ------------------------------------------------------------ deg -> d^{-1/2}
__global__ __launch_bounds__(256) void k_dinv(float* __restrict__ deg, int n) {
    int i = blockIdx.x * 256 + threadIdx.x;
    if (i < n) {
        float d = deg[i];
        deg[i] = (d > 0.0f) ? rsqrtf(d) : 0.0f;
    }
}

// ------------------------------------------- h = x @ W  (WMMA f32 16x16x4)
// 8 waves per block, one 16-row x 16-col output tile per wave; W in LDS.
__global__ __launch_bounds__(256) void k_gemm_xw(const float* __restrict__ x,
                                                 const float* __restrict__ W,
                                                 float* __restrict__ h, int n_tiles) {
    __shared__ float lw[F_IN * F_OUT];  // 8 KB, row-major [k][n]
    for (int i = threadIdx.x; i < F_IN * F_OUT; i += 256) lw[i] = W[i];
    __syncthreads();

    const int wave = threadIdx.x >> 5;
    const int lane = threadIdx.x & 31;
    const int tile = blockIdx.x * 8 + wave;
    if (tile >= n_tiles) return;  // wave-uniform

    const int half = lane >> 4;   // 0: K pair {k,k+1}; 1: {k+2,k+3}
    const int lm   = lane & 15;

    // A-frag source: row (tile*16 + lm), two consecutive K values per lane
    const float* xr = x + (size_t)(tile * 16 + lm) * F_IN + half * 2;

    v8f c = {};
#pragma unroll
    for (int k = 0; k < F_IN; k += 4) {
        v2f a = *(const v2f*)(xr + k);          // A 16x4: K=half*2, half*2+1
        const int kk = k + half * 2;
        v2f bf;                                  // B 4x16: row striped over lanes
        bf.x = lw[kk * F_OUT + lm];
        bf.y = lw[(kk + 1) * F_OUT + lm];
        c = __builtin_amdgcn_wmma_f32_16x16x4_f32(
            /*neg_a=*/false, a, /*neg_b=*/false, bf,
            /*c_mod=*/(short)0, c, /*reuse_a=*/false, /*reuse_b=*/false);
    }

    // D layout: VGPR r -> M = r + half*8, N = lm
    float* hp = h + ((size_t)tile * 16 + half * 8) * F_OUT + lm;
#pragma unroll
    for (int r = 0; r < 8; r++) hp[r * F_OUT] = c[r];
}

// --------------------------------------- out = b + dinv^2 * h  (self loops)
__global__ __launch_bounds__(256) void k_init_out(const float* __restrict__ dinv,
                                                  const float* __restrict__ h,
                                                  const float* __restrict__ bias,
                                                  float* __restrict__ out, int n) {
    int t = blockIdx.x * 256 + threadIdx.x;
    if (t >= n * F_OUT) return;
    int node = t >> 4;
    int f    = t & 15;
    float di = dinv[node];
    out[t] = bias[f] + di * di * h[t];
}

// ------------------- out[col] += h[row] * dinv[row]*w*dinv[col]  (messages)
// 16 consecutive lanes handle one edge -> 64B coalesced gather + atomics.
__global__ __launch_bounds__(256) void k_edge_scatter(const int* __restrict__ ei,
                                                      const float* __restrict__ ew,
                                                      const float* __restrict__ dinv,
                                                      const float* __restrict__ h,
                                                      float* __restrict__ out, int e) {
    long long t = (long long)blockIdx.x * 256 + threadIdx.x;
    if (t >= (long long)e * F_OUT) return;
    int edge = (int)(t >> 4);
    int f    = (int)(t & 15);
    int r = ei[edge];       // row = ei[0][edge]
    int c = ei[e + edge];   // col = ei[1][edge]
    float norm = dinv[r] * ew[edge] * dinv[c];
    atomicAdd(&out[(size_t)c * F_OUT + f], h[(size_t)r * F_OUT + f] * norm);
}

// ----------------------------------------------------------------- launcher
extern "C" void kernel_launch(void* const* d_in, const int* in_sizes, int n_in,
                              void* d_out, int out_size, void* d_ws, size_t ws_size,
                              hipStream_t stream) {
    const float* x  = (const float*)d_in[0];
    const int*   ei = (const int*)d_in[1];
    const float* ew = (const float*)d_in[2];
    const float* W  = (const float*)d_in[3];
    const float* b  = (const float*)d_in[4];
    float* out = (float*)d_out;

    const int n = in_sizes[0] / F_IN;  // 100000 nodes
    const int e = in_sizes[2];         // 3200000 edges

    // Workspace: [dinv: n floats][h: n*16 floats], h offset 256B-aligned
    char* ws = (char*)d_ws;
    float* dinv = (float*)ws;
    size_t hoff = (((size_t)n * sizeof(float)) + 255) & ~(size_t)255;
    float* h = (float*)(ws + hoff);

    const int nb_n = (n + 255) / 256;
    const int nb_e = (e + 255) / 256;

    k_init_deg<<<nb_n, 256, 0, stream>>>(dinv, n);
    k_deg_scatter<<<nb_e, 256, 0, stream>>>(ei, ew, dinv, e);
    k_dinv<<<nb_n, 256, 0, stream>>>(dinv, n);

    const int n_tiles = (n + 15) / 16;  // 6250 (n % 16 == 0 here)
    k_gemm_xw<<<(n_tiles + 7) / 8, 256, 0, stream>>>(x, W, h, n_tiles);

    const long long nf = (long long)n * F_OUT;
    k_init_out<<<(unsigned)((nf + 255) / 256), 256, 0, stream>>>(dinv, h, b, out, n);

    const long long ef = (long long)e * F_OUT;
    k_edge_scatter<<<(unsigned)((ef + 255) / 256), 256, 0, stream>>>(ei, ew, dinv, h, out, e);
}